// Net_Conv_14405320311021
// MI455X (gfx1250) — compile-verified
//
#include <hip/hip_runtime.h>
#include <hip/hip_bf16.h>

typedef __attribute__((ext_vector_type(2))) float v2f;
typedef __attribute__((ext_vector_type(8))) float v8f;

#define NEG_SLOPE 0.01f

// ---------------------------------------------------------------------------
// K1: init deg=1.0 (self-loop contribution) and t=0
// ---------------------------------------------------------------------------
__global__ void k_init(float* __restrict__ deg, float* __restrict__ t, int n) {
  int i = blockIdx.x * blockDim.x + threadIdx.x;
  if (i < n) { deg[i] = 1.0f; t[i] = 0.0f; }
}

// ---------------------------------------------------------------------------
// K2: degree accumulation over edge_orig targets (col = edge_orig[1])
// ---------------------------------------------------------------------------
__global__ void k_degree(const int* __restrict__ eo, float* __restrict__ deg,
                         long long E) {
  long long e = (long long)blockIdx.x * blockDim.x + threadIdx.x;
  if (e >= E) return;
  int c = eo[E + e];           // edge_orig[1][e]
  atomicAdd(&deg[c], 1.0f);
}

// ---------------------------------------------------------------------------
// K3: deg -> dinv = rsqrt(deg) (in place; deg >= 1 always due to self-loop)
// ---------------------------------------------------------------------------
__global__ void k_dinv(float* __restrict__ deg, int n) {
  int i = blockIdx.x * blockDim.x + threadIdx.x;
  if (i >= n) return;
  float d = deg[i];
  deg[i] = (d > 0.0f) ? rsqrtf(d) : 0.0f;
}

// ---------------------------------------------------------------------------
// K4: weighted scatter: t[col] += x[row] * dinv[row]
// ---------------------------------------------------------------------------
__global__ void k_scatter(const int* __restrict__ eo,
                          const float* __restrict__ x,
                          const float* __restrict__ dinv,
                          float* __restrict__ t, long long E) {
  long long e = (long long)blockIdx.x * blockDim.x + threadIdx.x;
  if (e >= E) return;
  int r = eo[e];               // edge_orig[0][e]
  int c = eo[E + e];           // edge_orig[1][e]
  atomicAdd(&t[c], x[r] * dinv[r]);
}

// ---------------------------------------------------------------------------
// K5: per-node h = leaky(s*W + b), then uv[16x4] = h[16x5] @ Wc[5x4] via
//     V_WMMA_F32_16X16X4_F32 (full-precision f32 WMMA), 16 nodes per wave.
//     Wc[j][0..1] = W2[j][0..1] (u part), Wc[j][2..3] = W2[5+j][0..1] (v part)
// ---------------------------------------------------------------------------
__global__ void k_node_uv_wmma(const float* __restrict__ x,
                               const float* __restrict__ dinv,
                               const float* __restrict__ t,
                               const float* __restrict__ W,   // 5
                               const float* __restrict__ b,   // 5
                               const float* __restrict__ W2,  // [10,2] row-major
                               float* __restrict__ uv,        // [n,4]
                               int n) {
  const int lane = threadIdx.x & 31;
  const int wave = threadIdx.x >> 5;
  const int wavesPerBlock = blockDim.x >> 5;
  const int tile = blockIdx.x * wavesPerBlock + wave;  // 16 nodes per tile
  const int base = tile * 16;
  if (base >= n) return;                               // wave-uniform exit

  // --- this lane's node scalar s and feature row h[0..4] ---
  const int m = lane & 15;
  const int node = base + m;
  float s = 0.0f;
  if (node < n) {
    float di = dinv[node];
    s = di * (t[node] + x[node] * di);
  }
  float h[5];
#pragma unroll
  for (int j = 0; j < 5; ++j) {
    float hv = s * W[j] + b[j];
    h[j] = (hv >= 0.0f) ? hv : NEG_SLOPE * hv;
  }

  // --- A matrices (16x4 f32 layout: lanes 0-15 K=0/1, lanes 16-31 K=2/3) ---
  const bool hi = (lane >= 16);
  v2f a1, a2;
  a1.x = hi ? h[2] : h[0];
  a1.y = hi ? h[3] : h[1];
  a2.x = hi ? 0.0f : h[4];     // second WMMA carries the K=4 term
  a2.y = 0.0f;

  // --- B matrices (4x16 f32, K striped over lane halves like A) ---
  const int col = lane & 15;
  v2f b1, b2m;
  b1.x = 0.0f; b1.y = 0.0f; b2m.x = 0.0f; b2m.y = 0.0f;
  if (col < 4) {
    const int k0 = hi ? 2 : 0;
    const int k1 = hi ? 3 : 1;
    // Wc[j][c] accessor
    auto wc = [&](int j, int c) -> float {
      return (c < 2) ? W2[j * 2 + c] : W2[(5 + j) * 2 + (c - 2)];
    };
    b1.x = wc(k0, col);
    b1.y = wc(k1, col);
    if (!hi) b2m.x = wc(4, col);
  }

  // --- D = A2*B2 + (A1*B1 + 0): two chained f32 WMMAs ---
  v8f c = {};
  c = __builtin_amdgcn_wmma_f32_16x16x4_f32(false, a1, false, b1,
                                            (short)0, c, false, false);
  c = __builtin_amdgcn_wmma_f32_16x16x4_f32(false, a2, false, b2m,
                                            (short)0, c, false, false);

  // --- write uv: lanes with (lane&15)<4 hold N=col; VGPR r holds M=r(+8) ---
  if (col < 4) {
    const int mOff = hi ? 8 : 0;
#pragma unroll
    for (int r = 0; r < 8; ++r) {
      int nd = base + mOff + r;
      if (nd < n) uv[(long long)nd * 4 + col] = c[r];
    }
  }
}

// ---------------------------------------------------------------------------
// K6: per-edge output: out[e,:] = uv[e1,0:2] + uv[e0,2:4] + b2
// ---------------------------------------------------------------------------
__global__ void k_edge_out(const int* __restrict__ ei,
                           const float* __restrict__ uv,
                           const float* __restrict__ b2,
                           float* __restrict__ out, long long E) {
  long long e = (long long)blockIdx.x * blockDim.x + threadIdx.x;
  if (e >= E) return;
  int i0 = ei[e];              // edge_index[0][e]
  int i1 = ei[E + e];          // edge_index[1][e]
  float2 u = *(const float2*)(uv + (long long)i1 * 4);      // u part of h[e1]
  float2 v = *(const float2*)(uv + (long long)i0 * 4 + 2);  // v part of h[e0]
  float2 o;
  o.x = u.x + v.x + b2[0];
  o.y = u.y + v.y + b2[1];
  *(float2*)(out + e * 2) = o;
}

// ---------------------------------------------------------------------------
extern "C" void kernel_launch(void* const* d_in, const int* in_sizes, int n_in,
                              void* d_out, int out_size, void* d_ws, size_t ws_size,
                              hipStream_t stream) {
  const float* x    = (const float*)d_in[0];   // [N,1]
  const int*   ei   = (const int*)d_in[1];     // [2,E]
  const int*   eo   = (const int*)d_in[2];     // [2,E]
  const float* W    = (const float*)d_in[3];   // [1,5]
  const float* b    = (const float*)d_in[4];   // [5]
  const float* W2   = (const float*)d_in[5];   // [10,2]
  const float* b2   = (const float*)d_in[6];   // [2]
  float*       out  = (float*)d_out;           // [E,2]

  const int       n = in_sizes[0];
  const long long E = (long long)in_sizes[1] / 2;

  // workspace carve: deg/dinv [n] | t [n] | uv [4n]
  float* deg = (float*)d_ws;
  float* t   = deg + n;
  float* uv  = t + n;

  const int BLK = 256;
  const int gN  = (n + BLK - 1) / BLK;
  const int gE  = (int)((E + BLK - 1) / BLK);
  const int tiles = (n + 15) / 16;                  // one wave per 16 nodes
  const int gT  = (tiles + (BLK / 32) - 1) / (BLK / 32);

  k_init<<<gN, BLK, 0, stream>>>(deg, t, n);
  k_degree<<<gE, BLK, 0, stream>>>(eo, deg, E);
  k_dinv<<<gN, BLK, 0, stream>>>(deg, n);
  k_scatter<<<gE, BLK, 0, stream>>>(eo, x, deg, t, E);
  k_node_uv_wmma<<<gT, BLK, 0, stream>>>(x, deg, t, W, b, W2, uv, n);
  k_edge_out<<<gE, BLK, 0, stream>>>(ei, uv, b2, out, E);
}